// MultiCategoryEncoder_6511170421583
// MI455X (gfx1250) — compile-verified
//
#include <hip/hip_runtime.h>

typedef __attribute__((ext_vector_type(2))) float v2f;
typedef __attribute__((ext_vector_type(8))) float v8f;

#define N_CLASSES 16
#define N_CAT     4096
#define OUTF      512
#define BS        4096

#define ROWS_PER_BLOCK 64          // padded class-segment granularity
#define KB             64          // k-chunk staged in LDS
#define ASTRIDE        68          // LDS row stride (floats): 68 % 64 == 4 -> conflict-free b64 reads
#define MAX_RBLOCKS    80          // ceil(4096/64) + 16 classes worst-case padding
#define SORT_CAP       5120       // >= 79 * 64

// d_ws int layout
#define WS_NB    0                 // [0]       number of row-blocks
#define WS_BC    1                 // [1..80]   class of each row-block
#define WS_SORT  128               // [128..]   padded sorted sample indices (-1 = pad)

// ---------------------------------------------------------------------------
// Kernel 1: counting sort of sample indices into 64-row-padded class segments.
// Single workgroup; fully initializes everything kernel 2 reads from ws.
// ---------------------------------------------------------------------------
__global__ void __launch_bounds__(256)
mce_sort_kernel(const int* __restrict__ classes, int* __restrict__ ws) {
  __shared__ int cnt[N_CLASSES];
  __shared__ int start[N_CLASSES];
  __shared__ int fill[N_CLASSES];
  const int tid = threadIdx.x;

  if (tid < N_CLASSES) { cnt[tid] = 0; fill[tid] = 0; }
  __syncthreads();

  for (int i = tid; i < BS; i += 256) atomicAdd(&cnt[classes[i]], 1);
  __syncthreads();

  if (tid == 0) {
    int run = 0, nb = 0;
    for (int c = 0; c < N_CLASSES; ++c) {
      start[c] = run;
      const int t = (cnt[c] + (ROWS_PER_BLOCK - 1)) / ROWS_PER_BLOCK;
      for (int j = 0; j < t; ++j) ws[WS_BC + nb + j] = c;
      nb  += t;
      run += t * ROWS_PER_BLOCK;
    }
    ws[WS_NB] = nb;
  }
  __syncthreads();

  for (int i = tid; i < SORT_CAP; i += 256) ws[WS_SORT + i] = -1;
  __syncthreads();

  for (int i = tid; i < BS; i += 256) {
    const int c   = classes[i];
    const int pos = start[c] + atomicAdd(&fill[c], 1);
    ws[WS_SORT + pos] = i;
  }
}

// ---------------------------------------------------------------------------
// Kernel 2: grouped GEMM with double-buffered LDS A staging.
// Block = 256 threads (8 waves) -> 64 rows x 128 cols of output.
// Wave w owns cols [w*16, w*16+16), accumulating four 16x16 fp32 C tiles
// via V_WMMA_F32_16X16X4_F32. One __syncthreads per 64-k chunk; next chunk's
// global loads are issued before the current chunk's WMMAs.
// ---------------------------------------------------------------------------
__global__ void __launch_bounds__(256)
mce_gemm_kernel(const float* __restrict__ batch,
                const float* __restrict__ W,
                const int*   __restrict__ ws,
                float*       __restrict__ out) {
  const int nb = ws[WS_NB];
  const int bx = blockIdx.x;
  if (bx >= nb) return;
  const int cls = ws[WS_BC + bx];

  __shared__ int   sidx[ROWS_PER_BLOCK];
  __shared__ float As[2][ROWS_PER_BLOCK * ASTRIDE];

  const int tid = threadIdx.x;
  if (tid < ROWS_PER_BLOCK) sidx[tid] = ws[WS_SORT + bx * ROWS_PER_BLOCK + tid];
  __syncthreads();

  const int wave  = tid >> 5;
  const int lane  = tid & 31;
  const int n     = lane & 15;            // row (A) / col (B) within 16
  const int khalf = (lane >> 4) << 1;     // lanes 16-31 hold k+2,k+3
  const int col   = blockIdx.y * 128 + wave * 16 + n;

  const float* bptr = W + (size_t)cls * OUTF * N_CAT + (size_t)col * N_CAT + khalf;

  // Cooperative A staging: 4 threads per row, 16 floats (4x float4) each.
  const int lr   = tid >> 2;
  const int seg  = (tid & 3) << 4;
  const int orig = sidx[lr];
  const float* ag   = batch + (size_t)(orig >= 0 ? orig : 0) * N_CAT + seg;
  const int    aoff = lr * ASTRIDE + seg;

  // Prologue: zero pad rows in BOTH buffers (never touched again); stage chunk 0.
  if (orig < 0) {
    const float4 z = {0.f, 0.f, 0.f, 0.f};
#pragma unroll
    for (int q = 0; q < 4; ++q) {
      *(float4*)(&As[0][aoff] + 4 * q) = z;
      *(float4*)(&As[1][aoff] + 4 * q) = z;
    }
  } else {
    const float4* g = (const float4*)ag;
    float4 x0 = g[0], x1 = g[1], x2 = g[2], x3 = g[3];
    *(float4*)(&As[0][aoff] +  0) = x0;
    *(float4*)(&As[0][aoff] +  4) = x1;
    *(float4*)(&As[0][aoff] +  8) = x2;
    *(float4*)(&As[0][aoff] + 12) = x3;
  }
  __syncthreads();

  v8f acc[4] = {};

  for (int k0 = 0; k0 < N_CAT; k0 += KB) {
    const int  buf       = (k0 / KB) & 1;
    const bool have_next = (k0 + KB) < N_CAT;

    // Issue next chunk's global loads early (latency hidden behind WMMAs).
    float4 y0, y1, y2, y3;
    if (have_next && orig >= 0) {
      const float4* g = (const float4*)(ag + k0 + KB);
      y0 = g[0]; y1 = g[1]; y2 = g[2]; y3 = g[3];
    }

    // Compute current chunk.
    const float* Ab = &As[buf][0];
    const float* bk = bptr + k0;
#pragma unroll
    for (int kk = 0; kk < KB; kk += 4) {
      const v2f b = *(const v2f*)(bk + kk);            // global_load_b64 (L2-resident W)
#pragma unroll
      for (int r = 0; r < 4; ++r) {
        const v2f a = *(const v2f*)(Ab + (r * 16 + n) * ASTRIDE + kk + khalf); // ds_load_b64
        acc[r] = __builtin_amdgcn_wmma_f32_16x16x4_f32(
            /*neg_a=*/false, a, /*neg_b=*/false, b,
            /*c_mod=*/(short)0, acc[r], /*reuse_a=*/false, /*reuse_b=*/false);
      }
    }

    // Commit next chunk into the alternate buffer.
    if (have_next && orig >= 0) {
      float* asl = &As[buf ^ 1][aoff];
      *(float4*)(asl +  0) = y0;
      *(float4*)(asl +  4) = y1;
      *(float4*)(asl +  8) = y2;
      *(float4*)(asl + 12) = y3;
    }
    __syncthreads();
  }

  // C layout: VGPR v -> row v (lanes 0-15) / row v+8 (lanes 16-31), col = lane&15.
  const int rowhalf = (lane >> 4) << 3;
#pragma unroll
  for (int r = 0; r < 4; ++r) {
#pragma unroll
    for (int v = 0; v < 8; ++v) {
      const int m = r * 16 + v + rowhalf;
      const int o = sidx[m];
      if (o >= 0) out[(size_t)o * OUTF + col] = acc[r][v];
    }
  }
}

// ---------------------------------------------------------------------------
extern "C" void kernel_launch(void* const* d_in, const int* in_sizes, int n_in,
                              void* d_out, int out_size, void* d_ws, size_t ws_size,
                              hipStream_t stream) {
  const float* batch   = (const float*)d_in[0];   // [4096, 4096] f32
  const int*   classes = (const int*)  d_in[1];   // [4096] i32
  const float* W       = (const float*)d_in[2];   // [16, 512, 4096] f32
  float*       out     = (float*)d_out;           // [4096, 512] f32
  int*         wsi     = (int*)d_ws;

  mce_sort_kernel<<<1, 256, 0, stream>>>(classes, wsi);
  mce_gemm_kernel<<<dim3(MAX_RBLOCKS, OUTF / 128), 256, 0, stream>>>(batch, W, wsi, out);
}